// Attention_47674136985799
// MI455X (gfx1250) — compile-verified
//
#include <hip/hip_runtime.h>
#include <math.h>

typedef __attribute__((ext_vector_type(2))) float v2f;
typedef __attribute__((ext_vector_type(8))) float v8f;

#define B_     64
#define H_     32
#define KVH_   8
#define G_     4
#define D_     128
#define BS_    128
#define NBPS_  16
#define S_     2048
#define NWAVES 4
#define TILE   16
#define SCALE_ 0.08838834764831845f  // 1/sqrt(128)

__device__ __forceinline__ float rmax16(float v) {
#pragma unroll
    for (int o = 1; o < 16; o <<= 1) v = fmaxf(v, __shfl_xor(v, o, 32));
    return v;
}
__device__ __forceinline__ float rsum16(float v) {
#pragma unroll
    for (int o = 1; o < 16; o <<= 1) v += __shfl_xor(v, o, 32);
    return v;
}

__device__ __forceinline__ v8f wmma4(v2f a, v2f b, v8f c) {
    // D = A(16x4) * B(4x16) + C, fp32, wave32
    return __builtin_amdgcn_wmma_f32_16x16x4_f32(
        /*neg_a=*/false, a, /*neg_b=*/false, b,
        /*c_mod=*/(short)0, c, /*reuse_a=*/false, /*reuse_b=*/false);
}

__launch_bounds__(128)
__global__ void pa_decode(const float* __restrict__ q,
                          const float* __restrict__ knew,
                          const float* __restrict__ vnew,
                          const float* __restrict__ kc,
                          const float* __restrict__ vc,
                          const int*   __restrict__ slot_mapping,  // derived, unused
                          const int*   __restrict__ block_tables,
                          const int*   __restrict__ ctx_lens,
                          float*       __restrict__ out) {
    __shared__ float sQ[16][D_];              // padded Q rows (heads 4..15 = 0)
    __shared__ float sP[NWAVES][16][TILE];    // per-wave P transpose bounce
    __shared__ float sO[NWAVES][G_][D_];      // per-wave partial outputs
    __shared__ float sM[NWAVES][G_];
    __shared__ float sL[NWAVES][G_];
    __shared__ int   sBT[NBPS_];

    const int bkv  = blockIdx.x;
    const int b    = bkv / KVH_;
    const int kv   = bkv % KVH_;
    const int tid  = threadIdx.x;
    const int lane = tid & 31;
    const int w    = tid >> 5;
    const int hi   = lane >> 4;   // lane half (selects K-pair of fragment)
    const int lh   = lane & 15;   // fragment row/column index

    if (tid < NBPS_) sBT[tid] = block_tables[b * NBPS_ + tid];
    for (int i = tid; i < 16 * D_; i += blockDim.x) {
        int row = i >> 7, d = i & (D_ - 1);
        sQ[row][d] = (row < G_) ? q[((b * H_) + kv * G_ + row) * D_ + d] : 0.f;
    }
    __syncthreads();

    const int  ctx    = ctx_lens[b];
    const int  last   = ctx - 1;
    const int  ntiles = (ctx + TILE - 1) / TILE;
    const long rowStr = (long)KVH_ * D_;            // floats between slots
    const float* kbase = kc + (long)kv * D_;
    const float* vbase = vc + (long)kv * D_;
    const float* knrow = knew + (long)(b * KVH_ + kv) * D_;
    const float* vnrow = vnew + (long)(b * KVH_ + kv) * D_;

    float m0 = -INFINITY, m1 = -INFINITY, m2 = -INFINITY, m3 = -INFINITY;
    float l0 = 0.f, l1 = 0.f, l2 = 0.f, l3 = 0.f;
    v8f acc[8] = {};   // O in WMMA C-layout, d-tiles j=0..7

    for (int t = w; t < ntiles; t += NWAVES) {
        const int p0   = t * TILE;
        const int posc = p0 + lh;                         // this lane's key column
        const long rowc = (long)sBT[posc >> 7] * BS_ + (posc & (BS_ - 1));
        const float* krow = (posc == last) ? knrow : (kbase + rowc * rowStr);

        // prefetch next tile's K row for this lane (global_prefetch_b8)
        if (t + NWAVES < ntiles) {
            int posn = (t + NWAVES) * TILE + lh;
            long rown = (long)sBT[posn >> 7] * BS_ + (posn & (BS_ - 1));
            const float* kp = (posn == last) ? knrow : (kbase + rown * rowStr);
            __builtin_prefetch(kp, 0, 0);
        }

        // ---- S = Q * K^T : 32 chained wmma_f32_16x16x4 over D ----
        v8f c = {};
#pragma unroll
        for (int ks = 0; ks < 32; ++ks) {
            const int dA = 4 * ks + 2 * hi;
            v2f a = { sQ[lh][dA], sQ[lh][dA + 1] };       // A: M=lh, K=dA..dA+1
            v2f bb = { krow[dA], krow[dA + 1] };          // B: N=lh, K=dA..dA+1
            c = wmma4(a, bb, c);
        }

        // ---- online softmax on the 4 valid head rows ----
        const bool pv = (posc < ctx);
        float sc[4], mn[4], sscale[4], p[4];
#pragma unroll
        for (int r = 0; r < 4; ++r) sc[r] = pv ? c[r] * SCALE_ : -INFINITY;
        float mo[4] = { m0, m1, m2, m3 };
#pragma unroll
        for (int r = 0; r < 4; ++r) {
            float tmax = rmax16(sc[r]);
            mn[r]     = fmaxf(mo[r], tmax);
            sscale[r] = __expf(mo[r] - mn[r]);
            p[r]      = pv ? __expf(sc[r] - mn[r]) : 0.f;
        }
        l0 = l0 * sscale[0] + rsum16(p[0]);
        l1 = l1 * sscale[1] + rsum16(p[1]);
        l2 = l2 * sscale[2] + rsum16(p[2]);
        l3 = l3 * sscale[3] + rsum16(p[3]);
        m0 = mn[0]; m1 = mn[1]; m2 = mn[2]; m3 = mn[3];

        // rescale accumulators (rows >=4 are padding; any scale is fine there)
#pragma unroll
        for (int j = 0; j < 8; ++j)
#pragma unroll
            for (int r = 0; r < 8; ++r) acc[j][r] *= sscale[r & 3];

        // ---- P: C-layout -> LDS -> A-layout ----
#pragma unroll
        for (int r = 0; r < 8; ++r) {
            float val = (!hi && r < 4) ? p[r] : 0.f;
            sP[w][hi * 8 + r][lh] = val;                  // rows 4..15 zeroed
        }

        // ---- O += P * V : 32 wmma over 8 d-tiles x 4 k-steps ----
#pragma unroll
        for (int s = 0; s < 4; ++s) {
            const int pr0 = p0 + 4 * s + 2 * hi;
            const int pr1 = pr0 + 1;
            long r0 = (long)sBT[pr0 >> 7] * BS_ + (pr0 & (BS_ - 1));
            long r1 = (long)sBT[pr1 >> 7] * BS_ + (pr1 & (BS_ - 1));
            const float* vr0 = (pr0 == last) ? vnrow : (vbase + r0 * rowStr);
            const float* vr1 = (pr1 == last) ? vnrow : (vbase + r1 * rowStr);
            v2f a = { sP[w][lh][4 * s + 2 * hi], sP[w][lh][4 * s + 2 * hi + 1] };
#pragma unroll
            for (int j = 0; j < 8; ++j) {
                const int dcol = 16 * j + lh;
                v2f bb = { vr0[dcol], vr1[dcol] };        // B: K=pos pair, N=dcol
                acc[j] = wmma4(a, bb, acc[j]);
            }
        }
    }

    // ---- publish per-wave partials ----
    if (lane == 0) {
        sM[w][0] = m0; sM[w][1] = m1; sM[w][2] = m2; sM[w][3] = m3;
        sL[w][0] = l0; sL[w][1] = l1; sL[w][2] = l2; sL[w][3] = l3;
    }
    if (!hi) {
#pragma unroll
        for (int j = 0; j < 8; ++j)
#pragma unroll
            for (int r = 0; r < 4; ++r) sO[w][r][16 * j + lh] = acc[j][r];
    }
    __syncthreads();

    // ---- cross-wave flash combine + store ----
    for (int i = tid; i < G_ * D_; i += blockDim.x) {
        const int hq = i >> 7, d = i & (D_ - 1);
        float M = -INFINITY;
#pragma unroll
        for (int w2 = 0; w2 < NWAVES; ++w2) M = fmaxf(M, sM[w2][hq]);
        float L = 0.f, o = 0.f;
#pragma unroll
        for (int w2 = 0; w2 < NWAVES; ++w2) {
            float wt = __expf(sM[w2][hq] - M);
            L += sL[w2][hq] * wt;
            o += sO[w2][hq][d] * wt;
        }
        out[((b * H_) + kv * G_ + hq) * D_ + d] = o / L;
    }
}

extern "C" void kernel_launch(void* const* d_in, const int* in_sizes, int n_in,
                              void* d_out, int out_size, void* d_ws, size_t ws_size,
                              hipStream_t stream) {
    const float* q    = (const float*)d_in[0];
    const float* k    = (const float*)d_in[1];
    const float* v    = (const float*)d_in[2];
    const float* kc   = (const float*)d_in[3];
    const float* vc   = (const float*)d_in[4];
    const int*   slot = (const int*)d_in[5];
    const int*   bt   = (const int*)d_in[6];
    const int*   cl   = (const int*)d_in[7];
    dim3 grid(B_ * KVH_), block(128);
    hipLaunchKernelGGL(pa_decode, grid, block, 0, stream,
                       q, k, v, kc, vc, slot, bt, cl, (float*)d_out);
}